// LinOSSLayer_10213432230020
// MI455X (gfx1250) — compile-verified
//
#include <hip/hip_runtime.h>
#include <hip/hip_bf16.h>

typedef __attribute__((ext_vector_type(2))) float v2f;
typedef __attribute__((ext_vector_type(8))) float v8f;

#define PDIM 256
#define HDIM 128
#define BSZ  4
#define LSEQ 4096
#define MROWS (BSZ * LSEQ)      // 16384
#define CHUNK 128
#define NCHUNK (LSEQ / CHUNK)   // 32
#define NT 4                    // N-subtiles per wave (16x64 tile)

// ---------------------------------------------------------------------------
// Stage 0: deinterleave the complex weight matrices once.
//   Bmat [P,H,2] -> bR, bI           [P,H]
//   Cmat [H,P,2] -> cR, cNI (= -Ci)  [H,P]
// Makes all GEMM B-operand loads contiguous (no per-k repack movs).
// ---------------------------------------------------------------------------
__global__ void linoss_split_w(const float* __restrict__ Bmat,
                               const float* __restrict__ Cmat,
                               float* __restrict__ bR, float* __restrict__ bI,
                               float* __restrict__ cR, float* __restrict__ cNI) {
    const int i = blockIdx.x * blockDim.x + threadIdx.x;   // 0 .. P*H-1
    bR[i]  =  Bmat[i * 2 + 0];
    bI[i]  =  Bmat[i * 2 + 1];
    cR[i]  =  Cmat[i * 2 + 0];
    cNI[i] = -Cmat[i * 2 + 1];
}

// ---------------------------------------------------------------------------
// Stage 1: Bu = u @ B^T  (real input, complex B) -> Bu_r, Bu_i  [M x P]
// One wave per 16(M) x 64(N) tile: A fragment loaded once per k-step and
// reused across 4 N-subtiles x {re,im} = 8 WMMAs.  K = HDIM, stepped by 4.
// ---------------------------------------------------------------------------
__global__ void linoss_gemm_bu(const float* __restrict__ u,      // [M, H]
                               const float* __restrict__ bR,     // [P, H]
                               const float* __restrict__ bI,     // [P, H]
                               float* __restrict__ buR,          // [M, P]
                               float* __restrict__ buI) {        // [M, P]
    const int lane    = threadIdx.x & 31;
    const int wave    = (blockIdx.x * (blockDim.x >> 5)) + (threadIdx.x >> 5);
    const int ngroups = PDIM / (16 * NT);          // 4
    const int ngroup  = wave % ngroups;
    const int tile_m  = wave / ngroups;            // 0..1023

    const int row = lane & 15;
    const int kb  = (lane >> 4) << 1;              // 0 or 2
    const int mg  = tile_m * 16 + row;             // global M row (A)

    v8f accR[NT] = {};
    v8f accI[NT] = {};

    const float* uRow = u + (size_t)mg * HDIM;
    const float* bRowR[NT];
    const float* bRowI[NT];
    #pragma unroll
    for (int nt = 0; nt < NT; ++nt) {
        const int ng = ngroup * (16 * NT) + nt * 16 + row;   // B row = p
        bRowR[nt] = bR + (size_t)ng * HDIM;
        bRowI[nt] = bI + (size_t)ng * HDIM;
    }

    for (int k0 = 0; k0 < HDIM; k0 += 4) {
        const v2f a = { uRow[k0 + kb + 0], uRow[k0 + kb + 1] };
        #pragma unroll
        for (int nt = 0; nt < NT; ++nt) {
            const v2f br = { bRowR[nt][k0 + kb + 0], bRowR[nt][k0 + kb + 1] };
            const v2f bi = { bRowI[nt][k0 + kb + 0], bRowI[nt][k0 + kb + 1] };
            accR[nt] = __builtin_amdgcn_wmma_f32_16x16x4_f32(
                false, a, false, br, (short)0, accR[nt], false, false);
            accI[nt] = __builtin_amdgcn_wmma_f32_16x16x4_f32(
                false, a, false, bi, (short)0, accI[nt], false, false);
        }
    }

    const int col = lane & 15;
    const int rhi = (lane >> 4) << 3;              // +8 for upper lane half
    #pragma unroll
    for (int nt = 0; nt < NT; ++nt) {
        #pragma unroll
        for (int v = 0; v < 8; ++v) {
            const size_t idx = (size_t)(tile_m * 16 + v + rhi) * PDIM
                             + ngroup * (16 * NT) + nt * 16 + col;
            buR[idx] = accR[nt][v];
            buI[idx] = accI[nt][v];
        }
    }
}

// ---------------------------------------------------------------------------
// Per-p transition coefficients (all real: A_diag, step are real).
// ---------------------------------------------------------------------------
__device__ __forceinline__ void linoss_coeffs(float a, float s,
                                              float& m11, float& m12,
                                              float& m21, float& m22) {
    const float schur = 1.0f / (1.0f + s * s * a);
    m11 = 1.0f - s * s * a * schur;
    m12 = -s * a * schur;
    m21 = s * schur;
    m22 = schur;
}

// ---------------------------------------------------------------------------
// Stage 2a: local scan of each chunk with zero initial state -> chunk finals.
// grid.x = BSZ * 2 * NCHUNK, block = PDIM threads (p = tid, coalesced).
// ---------------------------------------------------------------------------
__global__ void linoss_scan_chunks(const float* __restrict__ buR,
                                   const float* __restrict__ buI,
                                   const float* __restrict__ A_diag,
                                   const float* __restrict__ step,
                                   float* __restrict__ czf,   // [B*2*NC, P]
                                   float* __restrict__ cyf) {
    const int p    = threadIdx.x;
    const int c    = blockIdx.x % NCHUNK;
    const int comp = (blockIdx.x / NCHUNK) & 1;
    const int b    = blockIdx.x / (NCHUNK * 2);

    const float* bu = comp ? buI : buR;
    float m11, m12, m21, m22;
    const float s = step[p];
    linoss_coeffs(A_diag[p], s, m11, m12, m21, m22);

    float z = 0.0f, y = 0.0f;
    const size_t base = ((size_t)b * LSEQ + (size_t)c * CHUNK) * PDIM + p;
    for (int j = 0; j < CHUNK; ++j) {
        const float f = bu[base + (size_t)j * PDIM];
        const float t = z + s * f;
        const float zn = fmaf(m11, t, m12 * y);
        const float yn = fmaf(m21, t, m22 * y);
        z = zn; y = yn;
    }
    const size_t sidx = (size_t)blockIdx.x * PDIM + p;
    czf[sidx] = z;
    cyf[sidx] = y;
}

// ---------------------------------------------------------------------------
// Stage 2b: sequential combine across the 32 chunks using Mpow = M^CHUNK
// (7 squarings of the 2x2 matrix). Stores each chunk's incoming state and
// the final (z_f, y_f) into d_out.  grid = BSZ*2 blocks of PDIM threads.
// ---------------------------------------------------------------------------
__global__ void linoss_combine(const float* __restrict__ czf,
                               const float* __restrict__ cyf,
                               const float* __restrict__ A_diag,
                               const float* __restrict__ step,
                               float* __restrict__ czin,
                               float* __restrict__ cyin,
                               float* __restrict__ outF) {     // d_out base
    const int p    = threadIdx.x;
    const int comp = blockIdx.x & 1;
    const int b    = blockIdx.x >> 1;

    float m11, m12, m21, m22;
    linoss_coeffs(A_diag[p], step[p], m11, m12, m21, m22);

    // Mpow = M^128 by repeated squaring (2x2).
    float p11 = m11, p12 = m12, p21 = m21, p22 = m22;
    #pragma unroll
    for (int i = 0; i < 7; ++i) {
        const float q11 = p11 * p11 + p12 * p21;
        const float q12 = p11 * p12 + p12 * p22;
        const float q21 = p21 * p11 + p22 * p21;
        const float q22 = p21 * p12 + p22 * p22;
        p11 = q11; p12 = q12; p21 = q21; p22 = q22;
    }

    const size_t rowbase = (size_t)(b * 2 + comp) * NCHUNK;
    float z = 0.0f, y = 0.0f;
    for (int c = 0; c < NCHUNK; ++c) {
        const size_t sidx = (rowbase + c) * PDIM + p;
        czin[sidx] = z;
        cyin[sidx] = y;
        const float zf = czf[sidx];
        const float yf = cyf[sidx];
        const float zn = p11 * z + p12 * y + zf;
        const float yn = p21 * z + p22 * y + yf;
        z = zn; y = yn;
    }

    // Final states: complex interleaved [B, P, 2] after the main output.
    const size_t off = (size_t)MROWS * HDIM;
    outF[off + ((size_t)b * PDIM + p) * 2 + comp] = z;                          // z_f
    outF[off + (size_t)BSZ * PDIM * 2 + ((size_t)b * PDIM + p) * 2 + comp] = y; // y_f
}

// ---------------------------------------------------------------------------
// Stage 2c: re-scan each chunk from its correct incoming state, writing ys
// IN PLACE over Bu (each element read once then overwritten).
// ---------------------------------------------------------------------------
__global__ void linoss_fixup(float* __restrict__ buR,
                             float* __restrict__ buI,
                             const float* __restrict__ A_diag,
                             const float* __restrict__ step,
                             const float* __restrict__ czin,
                             const float* __restrict__ cyin) {
    const int p    = threadIdx.x;
    const int c    = blockIdx.x % NCHUNK;
    const int comp = (blockIdx.x / NCHUNK) & 1;
    const int b    = blockIdx.x / (NCHUNK * 2);

    float* bu = comp ? buI : buR;
    float m11, m12, m21, m22;
    const float s = step[p];
    linoss_coeffs(A_diag[p], s, m11, m12, m21, m22);

    const size_t sidx = (size_t)blockIdx.x * PDIM + p;
    float z = czin[sidx];
    float y = cyin[sidx];

    const size_t base = ((size_t)b * LSEQ + (size_t)c * CHUNK) * PDIM + p;
    for (int j = 0; j < CHUNK; ++j) {
        const size_t pos = base + (size_t)j * PDIM;
        const float f = bu[pos];
        const float t = z + s * f;
        const float zn = fmaf(m11, t, m12 * y);
        const float yn = fmaf(m21, t, m22 * y);
        z = zn; y = yn;
        bu[pos] = yn;           // ys (this component)
    }
}

// ---------------------------------------------------------------------------
// Stage 3: out = ys_r @ C_r^T + ys_i @ (-C_i)^T   [M x H]
// One wave per 16(M) x 64(N) tile; C_i pre-negated by the split pass.
// ---------------------------------------------------------------------------
__global__ void linoss_gemm_out(const float* __restrict__ ysR,   // [M, P]
                                const float* __restrict__ ysI,   // [M, P]
                                const float* __restrict__ cR,    // [H, P]
                                const float* __restrict__ cNI,   // [H, P]
                                float* __restrict__ outF) {      // [M, H]
    const int lane    = threadIdx.x & 31;
    const int wave    = (blockIdx.x * (blockDim.x >> 5)) + (threadIdx.x >> 5);
    const int ngroups = HDIM / (16 * NT);          // 2
    const int ngroup  = wave % ngroups;
    const int tile_m  = wave / ngroups;            // 0..1023

    const int row = lane & 15;
    const int kb  = (lane >> 4) << 1;
    const int mg  = tile_m * 16 + row;

    v8f acc[NT] = {};
    const float* arRow = ysR + (size_t)mg * PDIM;
    const float* aiRow = ysI + (size_t)mg * PDIM;
    const float* cRowR[NT];
    const float* cRowI[NT];
    #pragma unroll
    for (int nt = 0; nt < NT; ++nt) {
        const int ng = ngroup * (16 * NT) + nt * 16 + row;   // global h
        cRowR[nt] = cR  + (size_t)ng * PDIM;
        cRowI[nt] = cNI + (size_t)ng * PDIM;
    }

    for (int k0 = 0; k0 < PDIM; k0 += 4) {
        const v2f ar = { arRow[k0 + kb + 0], arRow[k0 + kb + 1] };
        const v2f ai = { aiRow[k0 + kb + 0], aiRow[k0 + kb + 1] };
        #pragma unroll
        for (int nt = 0; nt < NT; ++nt) {
            const v2f br = { cRowR[nt][k0 + kb + 0], cRowR[nt][k0 + kb + 1] };
            const v2f bi = { cRowI[nt][k0 + kb + 0], cRowI[nt][k0 + kb + 1] };
            acc[nt] = __builtin_amdgcn_wmma_f32_16x16x4_f32(
                false, ar, false, br, (short)0, acc[nt], false, false);
            acc[nt] = __builtin_amdgcn_wmma_f32_16x16x4_f32(
                false, ai, false, bi, (short)0, acc[nt], false, false);
        }
    }

    const int col = lane & 15;
    const int rhi = (lane >> 4) << 3;
    #pragma unroll
    for (int nt = 0; nt < NT; ++nt) {
        #pragma unroll
        for (int v = 0; v < 8; ++v) {
            outF[(size_t)(tile_m * 16 + v + rhi) * HDIM
                 + ngroup * (16 * NT) + nt * 16 + col] = acc[nt][v];
        }
    }
}

// ---------------------------------------------------------------------------
extern "C" void kernel_launch(void* const* d_in, const int* in_sizes, int n_in,
                              void* d_out, int out_size, void* d_ws, size_t ws_size,
                              hipStream_t stream) {
    const float* A_diag = (const float*)d_in[0];   // [P]
    const float* Bmat   = (const float*)d_in[1];   // [P,H,2]
    const float* Cmat   = (const float*)d_in[2];   // [H,P,2]
    const float* u      = (const float*)d_in[3];   // [B,L,H]
    const float* step   = (const float*)d_in[4];   // [P]
    float* outF         = (float*)d_out;

    // Workspace layout (floats)
    const size_t N1 = (size_t)MROWS * PDIM;            // 4,194,304
    const size_t N2 = (size_t)BSZ * 2 * NCHUNK * PDIM; // 65,536
    const size_t NW = (size_t)PDIM * HDIM;             // 32,768
    float* buR  = (float*)d_ws;
    float* buI  = buR + N1;
    float* czf  = buI + N1;
    float* cyf  = czf + N2;
    float* czin = cyf + N2;
    float* cyin = czin + N2;
    float* bR   = cyin + N2;
    float* bI   = bR + NW;
    float* cR   = bI + NW;
    float* cNI  = cR + NW;

    // Stage 0: deinterleave weights (P*H = 32768 elements).
    {
        dim3 blk(256);
        dim3 grd((PDIM * HDIM) / 256);
        linoss_split_w<<<grd, blk, 0, stream>>>(Bmat, Cmat, bR, bI, cR, cNI);
    }
    // Stage 1: Bu GEMM.  (M/16) * (P/64) = 4096 waves, 4 waves/block.
    {
        dim3 blk(128);
        dim3 grd((MROWS / 16) * (PDIM / (16 * NT)) / 4);
        linoss_gemm_bu<<<grd, blk, 0, stream>>>(u, bR, bI, buR, buI);
    }
    // Stage 2a: chunk-local scans.
    {
        dim3 blk(PDIM);
        dim3 grd(BSZ * 2 * NCHUNK);
        linoss_scan_chunks<<<grd, blk, 0, stream>>>(buR, buI, A_diag, step, czf, cyf);
    }
    // Stage 2b: combine chunk states + write final (z_f, y_f).
    {
        dim3 blk(PDIM);
        dim3 grd(BSZ * 2);
        linoss_combine<<<grd, blk, 0, stream>>>(czf, cyf, A_diag, step,
                                                czin, cyin, outF);
    }
    // Stage 2c: fixup re-scan, ys written in place over Bu.
    {
        dim3 blk(PDIM);
        dim3 grd(BSZ * 2 * NCHUNK);
        linoss_fixup<<<grd, blk, 0, stream>>>(buR, buI, A_diag, step, czin, cyin);
    }
    // Stage 3: output GEMM.  (M/16) * (H/64) = 2048 waves, 4 waves/block.
    {
        dim3 blk(128);
        dim3 grd((MROWS / 16) * (HDIM / (16 * NT)) / 4);
        linoss_gemm_out<<<grd, blk, 0, stream>>>(buR, buI, cR, cNI, outF);
    }
}